// Neural2DMinSumDecoder_13640816132467
// MI455X (gfx1250) — compile-verified
//
#include <hip/hip_runtime.h>
#include <stdint.h>

#define NN   8192          // variable nodes (2^13)
#define MM   4096          // check nodes
#define DVV  3             // variable degree
#define EE   (NN * DVV)    // 24576 edges
#define BB   512           // batch
#define TT   10            // BP iterations
#define THREADS 1024
#define VPT  (NN / THREADS)   // 8 variables per thread

// ---------------------------------------------------------------------------
// Edge e = l*N + i (variable v = edge_v[e], check c = i>>1) is stored
// check-major at slot (i>>1)*6 + 2*l + (i&1), so each check's 6 messages are
// 24 contiguous LDS bytes and ascending slot == ascending global edge index
// (preserves the reference argmin tie-break). Slot < 24576 fits in u16.
// ---------------------------------------------------------------------------
__global__ void ldpc_build_inverse(const int* __restrict__ edge_v,
                                   unsigned short* __restrict__ slot_of) {
    int e = blockIdx.x * blockDim.x + threadIdx.x;
    if (e >= EE) return;
    int l = e >> 13;            // layer
    int i = e & (NN - 1);       // position within layer
    int v = edge_v[e];
    int slot = (i >> 1) * 6 + 2 * l + (i & 1);
    slot_of[v * 3 + l] = (unsigned short)slot;
}

// LDS byte offset of a generic pointer into __shared__ (low 32 bits of the
// flat address are the LDS offset on gfx1250).
__device__ __forceinline__ unsigned lds_off(const void* p) {
    return (unsigned)(uintptr_t)p;
}

// gfx1250 async global->LDS, per-lane scatter, tracked with ASYNCcnt.
__device__ __forceinline__ void async_copy_b32(unsigned lds_addr,
                                               unsigned byte_off,
                                               unsigned long long gbase) {
    asm volatile("global_load_async_to_lds_b32 %0, %1, %2 offset:0"
                 :: "v"(lds_addr), "v"(byte_off), "s"(gbase) : "memory");
}

__device__ __forceinline__ void wait_async0() {
    asm volatile("s_wait_asynccnt 0" ::: "memory");
}

// ---------------------------------------------------------------------------
// One workgroup per codeword. Only the edge messages live in LDS (96 KB ->
// multiple WGs can co-reside on a 320 KB WGP). Each thread permanently owns
// 8 variables: their 24 edge slots + 8 channel LLRs stay in VGPRs.
// ---------------------------------------------------------------------------
extern "C" __global__ void __launch_bounds__(THREADS)
ldpc_decode_kernel(const float* __restrict__ llr,
                   const float* __restrict__ beta,
                   const float* __restrict__ alpha,
                   const unsigned short* __restrict__ slot_of,
                   float* __restrict__ out) {
    extern __shared__ char smem_raw[];
    float* msg = (float*)smem_raw;                  // EE floats = 96 KB

    const int tid = threadIdx.x;
    const int b   = blockIdx.x;
    const unsigned msg_base = lds_off(msg);

    // ---- per-thread persistent state: edge slots + channel LLRs ----
    int   eslot[VPT][3];
    float chllr[VPT];
    #pragma unroll
    for (int k = 0; k < VPT; ++k) {
        int v = tid + k * THREADS;
        chllr[k] = llr[(size_t)b * NN + v];
        #pragma unroll
        for (int l = 0; l < 3; ++l)
            eslot[k][l] = (int)slot_of[v * 3 + l];
    }

    // ---- init v2c = llr_e: async global->LDS scatter through the permutation
    {
        unsigned long long gllr =
            (unsigned long long)(uintptr_t)(llr + (size_t)b * NN);
        #pragma unroll
        for (int k = 0; k < VPT; ++k) {
            unsigned goff = (unsigned)((tid + k * THREADS) * 4);
            #pragma unroll
            for (int l = 0; l < 3; ++l)
                async_copy_b32(msg_base + (unsigned)eslot[k][l] * 4u, goff, gllr);
        }
        wait_async0();
    }
    __syncthreads();

    for (int t = 0; t < TT; ++t) {
        const float beta_t  = beta[t];
        const float alpha_t = alpha[t];

        // -------- check-node pass: min1/min2/first-argmin + sign parity ----
        // Check c's 6 messages are contiguous at msg[6c..6c+5]; three b64
        // accesses off one base (offsets 0/8/16), bank-conflict-free.
        for (int c = tid; c < MM; c += THREADS) {
            float* base = &msg[c * 6];
            float2 p0 = *(const float2*)(base + 0);
            float2 p1 = *(const float2*)(base + 2);
            float2 p2 = *(const float2*)(base + 4);
            float m[6] = {p0.x, p0.y, p1.x, p1.y, p2.x, p2.y};

            unsigned sb[6], parity = 0;
            float mag[6];
            #pragma unroll
            for (int k = 0; k < 6; ++k) {
                unsigned u = __float_as_uint(m[k]);
                sb[k]  = u >> 31;
                parity ^= sb[k];
                mag[k] = __uint_as_float(u & 0x7fffffffu);
            }
            float min1 = mag[0], min2 = 3.402823466e38f;
            int arg = 0;
            #pragma unroll
            for (int k = 1; k < 6; ++k) {
                if (mag[k] < min1) { min2 = min1; min1 = mag[k]; arg = k; }
                else               { min2 = fminf(min2, mag[k]); }
            }
            float c2v[6];
            #pragma unroll
            for (int k = 0; k < 6; ++k) {
                float val = beta_t * ((k == arg) ? min2 : min1);
                c2v[k] = (parity ^ sb[k]) ? -val : val;
            }
            *(float2*)(base + 0) = make_float2(c2v[0], c2v[1]);
            *(float2*)(base + 2) = make_float2(c2v[2], c2v[3]);
            *(float2*)(base + 4) = make_float2(c2v[4], c2v[5]);
        }
        __syncthreads();

        if (t < TT - 1) {
            // ---- variable-node pass: v2c = llr + alpha*(sum - own), all
            // indices/LLRs in registers, only message bytes touch LDS.
            #pragma unroll
            for (int k = 0; k < VPT; ++k) {
                int e0 = eslot[k][0], e1 = eslot[k][1], e2 = eslot[k][2];
                float c0 = msg[e0], c1 = msg[e1], c2 = msg[e2];
                float s  = c0 + c1 + c2;
                float ch = chllr[k];
                msg[e0] = ch + alpha_t * (s - c0);
                msg[e1] = ch + alpha_t * (s - c1);
                msg[e2] = ch + alpha_t * (s - c2);
            }
            __syncthreads();
        } else {
            // ---- posterior + hard decision ----
            #pragma unroll
            for (int k = 0; k < VPT; ++k) {
                int v = tid + k * THREADS;
                float p = chllr[k] + msg[eslot[k][0]] + msg[eslot[k][1]]
                                   + msg[eslot[k][2]];
                size_t o = (size_t)b * NN + (size_t)v;
                out[o]                   = (p < 0.0f) ? 1.0f : 0.0f; // bits
                out[(size_t)BB * NN + o] = p;                        // posterior
            }
        }
    }
}

// ---------------------------------------------------------------------------
// inputs: 0=llr [B*N] f32, 1=edge_v [E] i32, 2=edge_c [E] i32 (implicit in
// layout, unused), 3=beta [T] f32, 4=alpha [T] f32
// output: decoded_bits [B*N] then posterior [B*N], flat
// ---------------------------------------------------------------------------
extern "C" void kernel_launch(void* const* d_in, const int* in_sizes, int n_in,
                              void* d_out, int out_size, void* d_ws, size_t ws_size,
                              hipStream_t stream) {
    const float* llr    = (const float*)d_in[0];
    const int*   edge_v = (const int*)d_in[1];
    const float* beta   = (const float*)d_in[3];
    const float* alpha  = (const float*)d_in[4];

    unsigned short* slot_of = (unsigned short*)d_ws;   // N*3 u16 = 48 KB

    ldpc_build_inverse<<<(EE + 255) / 256, 256, 0, stream>>>(edge_v, slot_of);

    const size_t shmem = (size_t)EE * 4;               // 98304 B: msg only
    ldpc_decode_kernel<<<BB, THREADS, shmem, stream>>>(llr, beta, alpha,
                                                       slot_of, (float*)d_out);
}